// Net_1915555414429
// MI455X (gfx1250) — compile-verified
//
#include <hip/hip_runtime.h>
#include <hip/hip_bf16.h>
#include <math.h>

// Problem constants (from reference)
#define BB 2
#define TT 128
#define DD 384
#define HH 1024
#define CC 256
#define RR (BB * TT)          // 256 rows
#define NSHIFT (2 * DD - 1)   // 767
#define NELEM (RR * DD)       // 98304
#define TEMPER_F 10.0f
#define EPS_F 1e-6f

typedef __attribute__((ext_vector_type(2))) float v2f;
typedef __attribute__((ext_vector_type(8))) float v8f;

// ---------------------------------------------------------------------------
// Stage A: shift-scan + argmax + softmax attention + reverse shift -> x_ele
// One block (256 threads) per (b,t) row.
// ---------------------------------------------------------------------------
__device__ __forceinline__ void block_reduce_sum(float* rv, int tid) {
    for (int st = 128; st > 0; st >>= 1) {
        if (tid < st) rv[tid] += rv[tid + st];
        __syncthreads();
    }
}

__global__ void stage_a(const float* __restrict__ x, const float* __restrict__ y,
                        float* __restrict__ x_ele) {
    const int row = blockIdx.x;
    const int tid = threadIdx.x;
    __shared__ float xs[DD], ys[DD], xa[DD];
    __shared__ float rv[256];
    __shared__ int   ri[256];
    __shared__ float s_qn, s_dn, s_max, s_sum;
    __shared__ int   s_theta;

    const float* xr = x + row * DD;
    const float* yr = y + row * DD;
    for (int j = tid; j < DD; j += 256) { xs[j] = xr[j]; ys[j] = yr[j]; }
    __syncthreads();

    // qn = ||y||
    float acc = 0.f;
    for (int j = tid; j < DD; j += 256) acc += ys[j] * ys[j];
    rv[tid] = acc; __syncthreads();
    block_reduce_sum(rv, tid);
    if (tid == 0) s_qn = sqrtf(rv[0]);
    __syncthreads();
    const float qn = s_qn;

    // cosine score over 767 shifts, argmax with lowest-index tie break
    float best = -INFINITY; int bestS = 0x7fffffff;
    for (int s = tid; s < NSHIFT; s += 256) {
        const int off = s - (DD - 1);
        const int j0 = off < 0 ? -off : 0;
        const int j1 = off > 0 ? DD - off : DD;
        float dot = 0.f, kn2 = 0.f;
        for (int j = j0; j < j1; ++j) {
            const float xv = xs[j + off];
            dot += xv * ys[j];
            kn2 += xv * xv;
        }
        const float sim = dot / (qn * sqrtf(kn2) + EPS_F);
        if (sim > best || (sim == best && s < bestS)) { best = sim; bestS = s; }
    }
    rv[tid] = best; ri[tid] = bestS; __syncthreads();
    for (int st = 128; st > 0; st >>= 1) {
        if (tid < st) {
            const float ov = rv[tid + st]; const int oi = ri[tid + st];
            if (ov > rv[tid] || (ov == rv[tid] && oi < ri[tid])) { rv[tid] = ov; ri[tid] = oi; }
        }
        __syncthreads();
    }
    if (tid == 0) s_theta = ri[0];
    __syncthreads();
    const int theta = s_theta;
    const int off2 = theta - (DD - 1);   // x_opt[j] = xs[j+off2] when in range

    // dn = ||x_opt|| * ||y|| + eps
    acc = 0.f;
    for (int j = tid; j < DD; j += 256) {
        const int i = j + off2;
        const float v = (i >= 0 && i < DD) ? xs[i] : 0.f;
        acc += v * v;
    }
    rv[tid] = acc; __syncthreads();
    block_reduce_sum(rv, tid);
    if (tid == 0) s_dn = sqrtf(rv[0]) * qn + EPS_F;
    __syncthreads();
    const float dn = s_dn;

    // softmax over z[j] = x_opt[j]*y[j]/dn/TEMPER
    float lmax = -INFINITY;
    for (int j = tid; j < DD; j += 256) {
        const int i = j + off2;
        const float xo = (i >= 0 && i < DD) ? xs[i] : 0.f;
        const float z = (xo * ys[j] / dn) / TEMPER_F;
        lmax = fmaxf(lmax, z);
    }
    rv[tid] = lmax; __syncthreads();
    for (int st = 128; st > 0; st >>= 1) {
        if (tid < st) rv[tid] = fmaxf(rv[tid], rv[tid + st]);
        __syncthreads();
    }
    if (tid == 0) s_max = rv[0];
    __syncthreads();
    const float m = s_max;

    acc = 0.f;
    for (int j = tid; j < DD; j += 256) {
        const int i = j + off2;
        const float xo = (i >= 0 && i < DD) ? xs[i] : 0.f;
        const float z = (xo * ys[j] / dn) / TEMPER_F;
        acc += expf(z - m);
    }
    rv[tid] = acc; __syncthreads();
    block_reduce_sum(rv, tid);
    if (tid == 0) s_sum = rv[0];
    __syncthreads();
    const float sum = s_sum;

    // x_attn = x_opt * softmax(z)
    for (int j = tid; j < DD; j += 256) {
        const int i = j + off2;
        const float xo = (i >= 0 && i < DD) ? xs[i] : 0.f;
        const float z = (xo * ys[j] / dn) / TEMPER_F;
        xa[j] = xo * (expf(z - m) / sum);
    }
    __syncthreads();

    // reverse shift: x_ele[j] = x_attn[j + (D-1) - theta] when in range
    for (int j = tid; j < DD; j += 256) {
        const int i = j + (DD - 1) - theta;
        x_ele[row * DD + j] = (i >= 0 && i < DD) ? xa[i] : 0.f;
    }
}

// ---------------------------------------------------------------------------
// WMMA fp32 GEMM: D[M][N] = A[M][K] * B[K][N] + bias[N]
// One wave per 16x16 output tile; K stepped by 4 with v_wmma_f32_16x16x4_f32.
// A-frag: lanes 0-15 rows M=0..15 at (k0,k0+1); lanes 16-31 at (k0+2,k0+3).
// B-frag: mirrored (cols across lanes). C/D: VGPR r -> M = r (+8 for hi lanes).
// ---------------------------------------------------------------------------
template <int M, int K, int N>
__device__ __forceinline__ void wmma_gemm_tile(const float* __restrict__ A,
                                               const float* __restrict__ Bm,
                                               const float* __restrict__ bias,
                                               float* __restrict__ Dm,
                                               int wave, int lane) {
    const int nTiles = N / 16;
    const int nT = wave % nTiles;
    const int mT = wave / nTiles;
    const int row = mT * 16 + (lane & 15);
    const int col = nT * 16 + (lane & 15);
    const int khalf = (lane >> 4) * 2;   // 0 or 2

    v8f c = {};
    for (int k0 = 0; k0 < K; k0 += 4) {
        const int kk = k0 + khalf;
        v2f a, b;
        a.x = A[row * K + kk];
        a.y = A[row * K + kk + 1];
        b.x = Bm[kk * N + col];
        b.y = Bm[(kk + 1) * N + col];
        c = __builtin_amdgcn_wmma_f32_16x16x4_f32(false, a, false, b,
                                                  (short)0, c, false, false);
    }
    const float bv = bias[col];
    const int mbase = mT * 16 + ((lane >> 4) ? 8 : 0);
#pragma unroll
    for (int r2 = 0; r2 < 8; ++r2) {
        Dm[(mbase + r2) * N + col] = c[r2] + bv;
    }
}

__global__ void enc_gemm(const float* __restrict__ A, const float* __restrict__ Bm,
                         const float* __restrict__ bias, float* __restrict__ Dm) {
    const int wave = (blockIdx.x * blockDim.x + threadIdx.x) >> 5;  // 0..1023
    const int lane = threadIdx.x & 31;
    wmma_gemm_tile<RR, DD, HH>(A, Bm, bias, Dm, wave, lane);
}

__global__ void dec_gemm(const float* __restrict__ A, const float* __restrict__ Bm,
                         const float* __restrict__ bias, float* __restrict__ Dm) {
    const int wave = (blockIdx.x * blockDim.x + threadIdx.x) >> 5;  // 0..1535
    const int lane = threadIdx.x & 31;
    wmma_gemm_tile<4 * RR, HH, DD>(A, Bm, bias, Dm, wave, lane);
}

// ---------------------------------------------------------------------------
// Stage C: 4 iterations of suppress -> exact top-256 (bitonic sort of
// (value desc, index asc) 64-bit keys) -> masked h. One block per row.
// ---------------------------------------------------------------------------
__device__ __forceinline__ unsigned int fkey(float f) {
    unsigned int u = __float_as_uint(f);
    return (u & 0x80000000u) ? ~u : (u | 0x80000000u);  // ascending order key
}

__global__ void stage_c(const float* __restrict__ h0, float* __restrict__ hm) {
    const int r = blockIdx.x;
    const int tid = threadIdx.x;
    __shared__ float hs[HH];
    __shared__ unsigned long long keys[HH];
    __shared__ int sel[HH];

    for (int u = tid; u < HH; u += 256) { hs[u] = h0[r * HH + u]; sel[u] = 0; }
    __syncthreads();

    for (int it = 0; it < 4; ++it) {
        // build keys: descending value, ascending index on ties
        for (int u = tid; u < HH; u += 256) {
            const float v = sel[u] ? 1e-10f : hs[u];
            keys[u] = (((unsigned long long)(~fkey(v))) << 32) | (unsigned int)u;
        }
        __syncthreads();
        // bitonic sort ascending (keys[0..255] = top-256 after sort)
        for (int k = 2; k <= HH; k <<= 1) {
            for (int j = k >> 1; j > 0; j >>= 1) {
                for (int i = tid; i < HH; i += 256) {
                    const int ixj = i ^ j;
                    if (ixj > i) {
                        const bool up = ((i & k) == 0);
                        const unsigned long long a = keys[i], b = keys[ixj];
                        if ((a > b) == up) { keys[i] = b; keys[ixj] = a; }
                    }
                }
                __syncthreads();
            }
        }
        float* outp = hm + (size_t)(it * RR + r) * HH;
        for (int u = tid; u < HH; u += 256) outp[u] = 0.f;
        __syncthreads();
        // each of the 256 threads owns one of the top-256 entries
        const int idx = (int)(keys[tid] & 0xFFFFFFFFu);
        const float v = sel[idx] ? 1e-10f : hs[idx];
        outp[idx] = v;
        sel[idx] = 1;
        __syncthreads();
    }
}

// ---------------------------------------------------------------------------
// Loss reductions
// ---------------------------------------------------------------------------
__global__ void zero_acc(float* __restrict__ acc) {
    if (threadIdx.x < 5) acc[threadIdx.x] = 0.f;
}

__global__ void denom_k(const float* __restrict__ y, float* __restrict__ acc) {
    const int tid = threadIdx.x;
    const int e = blockIdx.x * 256 + tid;
    __shared__ float rv[256];
    rv[tid] = (y[e] != 0.0f) ? 1.f : 0.f;   // NELEM divisible by 256*384
    __syncthreads();
    block_reduce_sum(rv, tid);
    if (tid == 0) atomicAdd(&acc[0], rv[0]);
}

__global__ void loss_k(const float* __restrict__ out, const float* __restrict__ x,
                       const float* __restrict__ y, float* __restrict__ acc) {
    const int it = blockIdx.y;
    const int tid = threadIdx.x;
    const int e = blockIdx.x * 256 + tid;
    __shared__ float rv[256];
    float v = 0.f;
    if (y[e] != 0.0f) {
        const float d = out[(size_t)it * NELEM + e] - x[e];
        v = d * d;
    }
    rv[tid] = v;
    __syncthreads();
    block_reduce_sum(rv, tid);
    if (tid == 0) atomicAdd(&acc[1 + it], rv[0]);
}

__global__ void fin_k(const float* __restrict__ acc, float* __restrict__ losses) {
    if (threadIdx.x < 4) losses[threadIdx.x] = acc[1 + threadIdx.x] / acc[0];
}

// ---------------------------------------------------------------------------
// Launch
// ---------------------------------------------------------------------------
extern "C" void kernel_launch(void* const* d_in, const int* in_sizes, int n_in,
                              void* d_out, int out_size, void* d_ws, size_t ws_size,
                              hipStream_t stream) {
    const float* x     = (const float*)d_in[0];
    const float* y     = (const float*)d_in[1];
    const float* W_enc = (const float*)d_in[2];
    const float* b_enc = (const float*)d_in[3];
    const float* W_dec = (const float*)d_in[4];
    const float* b_dec = (const float*)d_in[5];
    float* out = (float*)d_out;
    float* ws  = (float*)d_ws;

    float* x_ele = ws;                       // 256*384   =   98304 floats
    float* h0    = ws + 98304;               // 256*1024  =  262144 floats
    float* hm    = ws + 360448;              // 4*256*1024= 1048576 floats
    float* acc   = ws + 1409024;             // 5 floats (denom + 4 losses)

    zero_acc<<<1, 32, 0, stream>>>(acc);
    stage_a<<<RR, 256, 0, stream>>>(x, y, x_ele);
    // enc: M=256, K=384, N=1024 -> 16*64 = 1024 tiles, 4 waves/block
    enc_gemm<<<256, 128, 0, stream>>>(x_ele, W_enc, b_enc, h0);
    stage_c<<<RR, 256, 0, stream>>>(h0, hm);
    // dec: M=1024, K=1024, N=384 -> 64*24 = 1536 tiles, 4 waves/block
    dec_gemm<<<384, 128, 0, stream>>>(hm, W_dec, b_dec, out);
    denom_k<<<NELEM / 256, 256, 0, stream>>>(y, acc);
    loss_k<<<dim3(NELEM / 256, 4), 256, 0, stream>>>(out, x, y, acc);
    fin_k<<<1, 32, 0, stream>>>(acc, out + 4 * NELEM);
}